// MeshInterpolateModule_31825707664043
// MI455X (gfx1250) — compile-verified
//
#include <hip/hip_runtime.h>
#include <stdint.h>

// Problem constants (from the reference): N,H,W,K,F,C = 4,512,512,1,10000,128
#define N_ 4
#define H_ 512
#define W_ 512
#define C_ 128

#define PIX_PER_BLK 32
#define PIX_STRIDE  (3 * C_ + 4)   // 388 floats = 1552 B: 16B-aligned rows, 2-way banked
#define THREADS     256

typedef int   v4i __attribute__((ext_vector_type(4)));
typedef float v4f __attribute__((ext_vector_type(4)));
typedef __attribute__((address_space(1))) v4i gv4i_t;   // global
typedef __attribute__((address_space(3))) v4i lv4i_t;   // LDS

__global__ __launch_bounds__(THREADS)
void mesh_interp_nchw_kernel(const int*   __restrict__ pix_to_face, // int32 per harness contract
                             const float* __restrict__ bary,        // (N,H,W,1,3)
                             const float* __restrict__ fm,          // (F,3,C)
                             float*       __restrict__ out)         // (N,C,H,W)
{
    __shared__ float tile[PIX_PER_BLK * PIX_STRIDE];   // ~48.5 KB

    const int tid  = threadIdx.x;
    const int lane = tid & 31;          // wave32
    const int wave = tid >> 5;          // 0..7

    const int bid  = blockIdx.x;
    const int wblk = bid & 15;                 // 16 tiles of 32 pixels per row
    const int h    = (bid >> 4) & (H_ - 1);
    const int n    = bid >> 13;
    const int w0   = wblk * PIX_PER_BLK;

    const long long pbase = ((long long)(n * H_ + h)) * W_ + w0;

    // Per-lane pixel == lane (reused as w in phase 2).
    const int  fidx  = pix_to_face[pbase + lane];
    const bool valid = (fidx >= 0);
    const int  fclmp = valid ? fidx : 0;

    // Unconditional loads (bary is defined for every pixel), then select:
    // keeps these as plain global_load_b32 + v_cndmask instead of exec-mask
    // branches. Zero weights reproduce the reference's mask since face-0
    // attributes are finite (0 * x == 0).
    const float* bp = bary + (pbase + lane) * 3;
    const float t0 = bp[0], t1 = bp[1], t2 = bp[2];
    const float b0 = valid ? t0 : 0.0f;
    const float b1 = valid ? t1 : 0.0f;
    const float b2 = valid ? t2 : 0.0f;

    // ---- Phase 1: async DMA 96 face rows (512 B each) global -> LDS ----
    // One address pair per pixel; the instruction's immediate offset advances
    // BOTH the global and LDS address by 512 B per vertex.
    #pragma unroll
    for (int pp = 0; pp < 4; ++pp) {
        const int p  = (wave << 2) + pp;          // pixel in tile, 0..31
        const int fp = __shfl(fclmp, p, 32);

        const float* g = fm + (size_t)fp * (3 * C_) + lane * 4;
        float*       l = tile + p * PIX_STRIDE + lane * 4;

#if __has_builtin(__builtin_amdgcn_global_load_async_to_lds_b128)
        __builtin_amdgcn_global_load_async_to_lds_b128((gv4i_t*)g, (lv4i_t*)l, 0,    0);
        __builtin_amdgcn_global_load_async_to_lds_b128((gv4i_t*)g, (lv4i_t*)l, 512,  0);
        __builtin_amdgcn_global_load_async_to_lds_b128((gv4i_t*)g, (lv4i_t*)l, 1024, 0);
#else
        unsigned ldsOff = (unsigned)(unsigned long long)
            ((__attribute__((address_space(3))) float*)l);
        asm volatile("global_load_async_to_lds_b128 %0, %1, off\n\t"
                     "global_load_async_to_lds_b128 %0, %1, off offset:512\n\t"
                     "global_load_async_to_lds_b128 %0, %1, off offset:1024"
                     :: "v"(ldsOff), "v"(g) : "memory");
#endif
    }

#if __has_builtin(__builtin_amdgcn_s_wait_asynccnt)
    __builtin_amdgcn_s_wait_asynccnt(0);
#else
    asm volatile("s_wait_asynccnt 0x0" ::: "memory");
#endif
    __syncthreads();

    // ---- Phase 2: 3-term FMA from LDS (float4), NCHW-transposed stores ----
    // Wave owns a contiguous 16-channel slab; lane carries w, so every store
    // is a contiguous 128 B burst. Channel-plane stride is 1 MiB -> stores use
    // the 24-bit immediate offset (two bases cover planes 0..7 and 8..15).
    const size_t plane = (size_t)H_ * W_;        // 262144 floats = 1 MiB
    const int    c0    = wave << 4;              // 16 channels per wave
    const float* lptr  = tile + lane * PIX_STRIDE + c0;

    float* ob0 = out + (((size_t)(n * C_ + c0) * H_ + h) * (size_t)W_ + w0 + lane);
    float* ob8 = ob0 + 8 * plane;

    #pragma unroll
    for (int i = 0; i < 4; ++i) {
        const v4f av = *(const v4f*)(lptr +          i * 4);
        const v4f bv = *(const v4f*)(lptr + C_     + i * 4);
        const v4f cv = *(const v4f*)(lptr + 2 * C_ + i * 4);
        #pragma unroll
        for (int j = 0; j < 4; ++j) {
            float r = b0 * av[j];
            r = __builtin_fmaf(b1, bv[j], r);
            r = __builtin_fmaf(b2, cv[j], r);
            const int m = i * 4 + j;             // channel within slab, 0..15
            if (m < 8) ob0[(size_t)m       * plane] = r;
            else       ob8[(size_t)(m - 8) * plane] = r;
        }
    }
}

extern "C" void kernel_launch(void* const* d_in, const int* in_sizes, int n_in,
                              void* d_out, int out_size, void* d_ws, size_t ws_size,
                              hipStream_t stream) {
    (void)in_sizes; (void)n_in; (void)out_size; (void)d_ws; (void)ws_size;

    const int*   pix  = (const int*)  d_in[0];   // pix_to_face (N,H,W,1)
    const float* bary = (const float*)d_in[1];   // bary_coords (N,H,W,1,3)
    const float* fm   = (const float*)d_in[2];   // face_memory (F,3,C)
    float*       outp = (float*)      d_out;     // (N,C,H,W)

    const int blocks = N_ * H_ * (W_ / PIX_PER_BLK);  // 32768
    mesh_interp_nchw_kernel<<<blocks, THREADS, 0, stream>>>(pix, bary, fm, outp);
}